// BertNer_44306882625826
// MI455X (gfx1250) — compile-verified
//
#include <hip/hip_runtime.h>
#include <math.h>

#define B_ 128
#define S_ 512
#define H_ 1024
#define L_ 9

typedef __attribute__((ext_vector_type(2))) float v2f;
typedef __attribute__((ext_vector_type(8))) float v8f;

// ---------------------------------------------------------------------------
// Kernel 1: per-batch stable compaction. idx[b][j] = source token index of the
// j-th valid token (order preserved), or -1 for pad slots.
// ---------------------------------------------------------------------------
__global__ void __launch_bounds__(S_) compact_idx_kernel(const int* __restrict__ valid,
                                                         int* __restrict__ idx) {
  __shared__ int scan[2][S_];
  __shared__ int slot[S_];
  const int s = threadIdx.x;
  const int b = blockIdx.x;
  const int v = valid[b * S_ + s];
  scan[0][s] = v;
  slot[s] = -1;
  __syncthreads();
  int cur = 0;
  for (int off = 1; off < S_; off <<= 1) {
    int t = scan[cur][s];
    if (s >= off) t += scan[cur][s - off];
    scan[cur ^ 1][s] = t;
    cur ^= 1;
    __syncthreads();
  }
  const int excl = scan[cur][s] - v;   // exclusive prefix of valid count
  if (v) slot[excl] = s;
  __syncthreads();
  idx[b * S_ + s] = slot[s];
}

// ---------------------------------------------------------------------------
// Kernel 2: gathered GEMM [16 x 1024] x [1024 x 16(pad)] per wave via
// V_WMMA_F32_16X16X4_F32, + bias + softmax over the 9 real classes.
// Block = 128 threads (4 waves); each wave owns one 16-token output tile.
// ---------------------------------------------------------------------------
__global__ void __launch_bounds__(128) ner_head_kernel(const float* __restrict__ seq,
                                                       const float* __restrict__ W,
                                                       const float* __restrict__ bias,
                                                       const int* __restrict__ idx,
                                                       float* __restrict__ out) {
  // W padded to [1024][16] f32 in LDS (zero columns for n >= 9): 64 KB.
  __shared__ float Wlds[H_ * 16];
  const int tid = threadIdx.x;
  for (int i = tid; i < H_ * 16; i += 128) {
    const int k = i >> 4;
    const int n = i & 15;
    Wlds[i] = (n < L_) ? W[k * L_ + n] : 0.0f;
  }
  __syncthreads();

  const int b    = blockIdx.y;
  const int wave = tid >> 5;
  const int lane = tid & 31;
  const int row  = lane & 15;        // M (A-frag) / N (B-frag, C columns)
  const int hi   = lane >> 4;        // 0: K=k0..k0+1, 1: K=k0+2..k0+3
  const int ko   = hi * 2;
  const int j0   = blockIdx.x * 64 + wave * 16;  // output-slot tile base

  // Gather: this lane streams source row idx[b][j0+row]; pad rows -> masked 0.
  int src = idx[b * S_ + j0 + row];
  const float mval = (src >= 0) ? 1.0f : 0.0f;
  src = (src >= 0) ? src : 0;
  const float* aBase = seq + ((size_t)b * S_ + (size_t)src) * H_ + ko;

  v8f acc = {};
#pragma unroll 8
  for (int k0 = 0; k0 < H_; k0 += 4) {
    v2f a = *(const v2f*)(aBase + k0);   // A[M=row][k0+ko .. k0+ko+1]
    a.x *= mval;
    a.y *= mval;
    v2f bb;                              // B[K][N=row]
    bb.x = Wlds[(k0 + ko) * 16 + row];
    bb.y = Wlds[(k0 + ko + 1) * 16 + row];
    acc = __builtin_amdgcn_wmma_f32_16x16x4_f32(
        /*neg_a=*/false, a, /*neg_b=*/false, bb,
        /*c_mod=*/(short)0, acc, /*reuse_a=*/false, /*reuse_b=*/false);
  }

  // Epilogue: bias (-inf on pad columns so exp -> 0), softmax over 16 lanes.
  const float bn = (row < L_) ? bias[row] : -INFINITY;
#pragma unroll
  for (int v = 0; v < 8; ++v) {
    const int m = v + hi * 8;            // C layout: lanes 16-31 hold M = v+8
    float x = acc[v] + bn;
    float mx = x;
    mx = fmaxf(mx, __shfl_xor(mx, 1, 16));
    mx = fmaxf(mx, __shfl_xor(mx, 2, 16));
    mx = fmaxf(mx, __shfl_xor(mx, 4, 16));
    mx = fmaxf(mx, __shfl_xor(mx, 8, 16));
    float e = __expf(x - mx);
    float sm = e;
    sm += __shfl_xor(sm, 1, 16);
    sm += __shfl_xor(sm, 2, 16);
    sm += __shfl_xor(sm, 4, 16);
    sm += __shfl_xor(sm, 8, 16);
    if (row < L_) {
      out[((size_t)b * S_ + (size_t)(j0 + m)) * L_ + row] = e / sm;
    }
  }
}

// ---------------------------------------------------------------------------
extern "C" void kernel_launch(void* const* d_in, const int* in_sizes, int n_in,
                              void* d_out, int out_size, void* d_ws, size_t ws_size,
                              hipStream_t stream) {
  const float* seq   = (const float*)d_in[0];   // [B,S,H]
  const float* W     = (const float*)d_in[1];   // [H,L]
  const float* bias  = (const float*)d_in[2];   // [L]
  const int*   valid = (const int*)d_in[3];     // [B,S]
  float*       out   = (float*)d_out;           // [B,S,L]
  int*         idx   = (int*)d_ws;              // [B,S] compaction indices

  compact_idx_kernel<<<dim3(B_), dim3(S_), 0, stream>>>(valid, idx);
  ner_head_kernel<<<dim3(S_ / 64, B_), dim3(128), 0, stream>>>(seq, W, bias, idx, out);
}